// PagedKVCache_79087527789038
// MI455X (gfx1250) — compile-verified
//
#include <hip/hip_runtime.h>
#include <cstdint>

// Problem constants (match reference):
// k_cache/v_cache: [4096, 8, 16, 128] f32 ; k_val/v_val: [32, 512, 8, 128] f32
// slot_mapping: [32, 512] int ; out: [2, 4096, 8, 16, 128] f32
#define NPAGES 4096
#define NHEADS 8
#define PAGESZ 16
#define HEADD  128
#define NTOK   (32 * 512)

typedef float v4f __attribute__((ext_vector_type(4)));
typedef int   g4i __attribute__((vector_size(16)));   // type the async builtins want

#define GLOBAL_AS __attribute__((address_space(1)))
#define LDS_AS    __attribute__((address_space(3)))

// Temporal hint: NT (non-temporal) for both loads and stores per CDNA5 TH
// encoding tables — the 1 GB copy has zero reuse and is 5x the 192 MB L2.
#define CPOL_NT 1

#if __has_builtin(__builtin_amdgcn_global_load_async_to_lds_b128) && \
    __has_builtin(__builtin_amdgcn_global_store_async_from_lds_b128)
#define HAVE_ASYNC_LDS 1
#else
#define HAVE_ASYNC_LDS 0
#endif

__device__ __forceinline__ void wait_async0() {
#if __has_builtin(__builtin_amdgcn_s_wait_asynccnt)
  __builtin_amdgcn_s_wait_asynccnt(0);
#else
  asm volatile("s_wait_asynccnt 0" ::: "memory");
#endif
}

// ---------------------------------------------------------------------------
// Kernel 1: stream both caches into the output buffer.
//   out[0 .. n4)      = k_cache
//   out[n4 .. 2*n4)   = v_cache        (n4 in float4 units)
// HBM-bound: 512 MB read + 512 MB write. Uses gfx1250 async LDS staging
// (GLOBAL_LOAD_ASYNC_TO_LDS_B128 / GLOBAL_STORE_ASYNC_FROM_LDS_B128, tracked
// by ASYNCcnt) — no VGPR round-trip for the bulk copy.
// All offsets fit in 32 bits (max 256 MB) -> unsigned indexing so the
// compiler can use saddr + 32-bit voffset addressing.
// ---------------------------------------------------------------------------
__global__ __launch_bounds__(256) void pagedkv_copy_caches(
    const v4f* __restrict__ kc, const v4f* __restrict__ vc,
    v4f* __restrict__ out, unsigned n4) {
  unsigned i = blockIdx.x * blockDim.x + threadIdx.x;
  const unsigned stride = gridDim.x * blockDim.x;
  v4f* __restrict__ outk = out;
  v4f* __restrict__ outv = out + n4;

#if HAVE_ASYNC_LDS
  __shared__ v4f stage[512];                       // 256 lanes x 16B x {k,v}
  LDS_AS g4i* lk = (LDS_AS g4i*)&stage[threadIdx.x];
  LDS_AS g4i* lv = (LDS_AS g4i*)&stage[256 + threadIdx.x];
  for (; i < n4; i += stride) {
    __builtin_amdgcn_global_load_async_to_lds_b128(
        (GLOBAL_AS g4i*)(kc + i), lk, 0, CPOL_NT);
    __builtin_amdgcn_global_load_async_to_lds_b128(
        (GLOBAL_AS g4i*)(vc + i), lv, 0, CPOL_NT);
    wait_async0();                                  // LDS side is unordered
    __builtin_amdgcn_global_store_async_from_lds_b128(
        (GLOBAL_AS g4i*)(outk + i), lk, 0, CPOL_NT);
    __builtin_amdgcn_global_store_async_from_lds_b128(
        (GLOBAL_AS g4i*)(outv + i), lv, 0, CPOL_NT);
    wait_async0();                                  // before LDS reuse
  }
#else
  for (; i < n4; i += stride) {
    __builtin_prefetch(kc + i + stride, 0, 0);      // global_prefetch_b8
    __builtin_prefetch(vc + i + stride, 0, 0);
    v4f a = __builtin_nontemporal_load(kc + i);     // b128, th:NT (no reuse)
    v4f b = __builtin_nontemporal_load(vc + i);
    __builtin_nontemporal_store(a, outk + i);
    __builtin_nontemporal_store(b, outv + i);
  }
#endif
}

// ---------------------------------------------------------------------------
// Kernel 2: scatter k_val/v_val rows into the copied caches.
// One wave32 per (token, head) row: D=128 floats / 32 lanes = one float4 per
// lane -> each 512B row is a fully coalesced b128 store. slot is wave-uniform
// (scalarizable). All element offsets < 2^25 -> 32-bit indexing.
// ---------------------------------------------------------------------------
__global__ __launch_bounds__(256) void pagedkv_scatter(
    const v4f* __restrict__ kv, const v4f* __restrict__ vv,
    const int* __restrict__ slots, v4f* __restrict__ out,
    unsigned n4, unsigned n_rows) {
  unsigned gid  = blockIdx.x * blockDim.x + threadIdx.x;
  unsigned row  = gid >> 5;       // 32 lanes (wave32) per row
  unsigned lane = gid & 31;
  if (row >= n_rows) return;

  unsigned tok = row >> 3;        // NHEADS == 8
  unsigned h   = row & 7;
  unsigned slot = (unsigned)slots[tok];  // wave-uniform -> scalarized
  unsigned page = slot >> 4;      // PAGESZ == 16
  unsigned off  = slot & 15;

  // src: [tok, h, d] row-major; lane covers d = lane*4 .. lane*4+3
  unsigned src = row * (HEADD / 4) + lane;
  // dst: [page, h, off, d]
  unsigned dst = (((page * NHEADS + h) * PAGESZ + off) * (HEADD / 4)) + lane;

  v4f k = __builtin_nontemporal_load(kv + src);     // streamed, no reuse
  v4f v = __builtin_nontemporal_load(vv + src);
  __builtin_nontemporal_store(k, out + dst);        // scattered, no reuse
  __builtin_nontemporal_store(v, out + n4 + dst);
}

extern "C" void kernel_launch(void* const* d_in, const int* in_sizes, int n_in,
                              void* d_out, int out_size, void* d_ws,
                              size_t ws_size, hipStream_t stream) {
  const v4f*  kc    = (const v4f*)d_in[0];   // k_cache  [4096,8,16,128] f32
  const v4f*  vc    = (const v4f*)d_in[1];   // v_cache
  const v4f*  kv    = (const v4f*)d_in[2];   // k_val    [32,512,8,128] f32
  const v4f*  vv    = (const v4f*)d_in[3];   // v_val
  const int*  slots = (const int*)d_in[4];   // slot_mapping [32,512]
  v4f*        out   = (v4f*)d_out;           // [2,4096,8,16,128] f32

  const unsigned n4 = (unsigned)((long long)NPAGES * NHEADS * PAGESZ * HEADD / 4);

  // 1) copy caches -> out (grid-stride; 8192*256 threads, 8 iters each)
  pagedkv_copy_caches<<<8192, 256, 0, stream>>>(kc, vc, out, n4);

  // 2) scatter token rows over the copy
  const unsigned n_rows  = NTOK * NHEADS;     // 131072 waves
  const int      threads = 256;
  const int      blocks  = (int)((n_rows * 32u + threads - 1) / threads);  // 16384
  pagedkv_scatter<<<blocks, threads, 0, stream>>>(kv, vv, slots, out, n4,
                                                  n_rows);
}